// BeamSearchDecoder_40518721470927
// MI455X (gfx1250) — compile-verified
//
#include <hip/hip_runtime.h>
#include <hip/hip_bf16.h>
#include <math.h>

#define HD   512
#define LD   2
#define BW   5
#define VSZ  32000
#define TDEC 64
#define NEGF (-1000000000.0f)

typedef float v2f __attribute__((ext_vector_type(2)));
typedef float v8f __attribute__((ext_vector_type(8)));

__device__ __forceinline__ float wave_sum(float v) {
#pragma unroll
    for (int off = 16; off; off >>= 1) v += __shfl_xor(v, off, 32);
    return v;
}
__device__ __forceinline__ float sigm(float x) { return 1.0f / (1.0f + expf(-x)); }

__global__ void k_zero(float* p, int n) {
    int i = blockIdx.x * blockDim.x + threadIdx.x;
    if (i < n) p[i] = 0.0f;
}

// Fused GRU cell: one wave per (beam b, output j). Computes all six 512-long
// dot products, wave-reduces, applies gate math on lane 0.
// x row is optionally gathered via x_index (token embedding lookup),
// h row optionally gathered via parent (beam-search hidden re-ordering).
__global__ void k_gru(const float* __restrict__ x, const int* __restrict__ xind,
                      const float* __restrict__ h, const int* __restrict__ par,
                      const float* __restrict__ Wi, const float* __restrict__ Wh,
                      const float* __restrict__ bi, const float* __restrict__ bh,
                      float* __restrict__ hout, float* __restrict__ eout, int B) {
    int wid  = (blockIdx.x * blockDim.x + threadIdx.x) >> 5;
    int lane = threadIdx.x & 31;
    if (wid >= B * HD) return;
    int b = wid >> 9, j = wid & (HD - 1);
    const float* xr = x + (size_t)(xind ? xind[b] : b) * HD;
    const float* hr = h + (size_t)(par ? par[b] : b) * HD;
    const float *wir = Wi + (size_t)j * HD, *wiz = Wi + (size_t)(j + HD) * HD,
                *win = Wi + (size_t)(j + 2 * HD) * HD;
    const float *whr = Wh + (size_t)j * HD, *whz = Wh + (size_t)(j + HD) * HD,
                *whn = Wh + (size_t)(j + 2 * HD) * HD;
    float air = 0.f, aiz = 0.f, ain = 0.f, ahr = 0.f, ahz = 0.f, ahn = 0.f;
    for (int k = lane; k < HD; k += 32) {
        float xv = xr[k], hv = hr[k];
        air = fmaf(xv, wir[k], air); aiz = fmaf(xv, wiz[k], aiz); ain = fmaf(xv, win[k], ain);
        ahr = fmaf(hv, whr[k], ahr); ahz = fmaf(hv, whz[k], ahz); ahn = fmaf(hv, whn[k], ahn);
    }
    air = wave_sum(air); aiz = wave_sum(aiz); ain = wave_sum(ain);
    ahr = wave_sum(ahr); ahz = wave_sum(ahz); ahn = wave_sum(ahn);
    if (lane == 0) {
        float r = sigm(air + bi[j] + ahr + bh[j]);
        float z = sigm(aiz + bi[j + HD] + ahz + bh[j + HD]);
        float n = tanhf(ain + bi[j + 2 * HD] + r * (ahn + bh[j + 2 * HD]));
        float hp = (1.0f - z) * n + z * hr[j];
        hout[(size_t)b * HD + j] = hp;
        if (eout) eout[(size_t)b * HD + j] = hp;
    }
}

// Attention scores: e[b][t] = rnn[b] . enc_out[t], one wave per (b,t).
__global__ void k_attn(const float* __restrict__ rnn, const float* __restrict__ enc_out,
                       float* __restrict__ e, int T) {
    int wid = (blockIdx.x * blockDim.x + threadIdx.x) >> 5;
    int lane = threadIdx.x & 31;
    if (wid >= BW * T) return;
    int b = wid / T, t = wid - b * T;
    const float* r = rnn + (size_t)b * HD;
    const float* o = enc_out + (size_t)t * HD;
    float a = 0.f;
    for (int k = lane; k < HD; k += 32) a = fmaf(r[k], o[k], a);
    a = wave_sum(a);
    if (lane == 0) e[b * T + t] = a;
}

// In-place softmax along T (T power of two, <=256). One block per beam.
__global__ void k_softmax(float* __restrict__ e, int T) {
    int b = blockIdx.x, t = threadIdx.x;
    __shared__ float sm[256];
    float v = e[b * T + t];
    sm[t] = v; __syncthreads();
    for (int s = T >> 1; s > 0; s >>= 1) { if (t < s) sm[t] = fmaxf(sm[t], sm[t + s]); __syncthreads(); }
    float mx = sm[0]; __syncthreads();
    float ex = expf(v - mx);
    sm[t] = ex; __syncthreads();
    for (int s = T >> 1; s > 0; s >>= 1) { if (t < s) sm[t] += sm[t + s]; __syncthreads(); }
    e[b * T + t] = ex / sm[0];
}

// ctx[b][h] = sum_t a[b][t] * enc_out[t][h]
__global__ void k_ctx(const float* __restrict__ a, const float* __restrict__ enc_out,
                      float* __restrict__ ctx, int T) {
    int i = blockIdx.x * blockDim.x + threadIdx.x;
    if (i >= BW * HD) return;
    int b = i / HD, hh = i - b * HD;
    float acc = 0.f;
    for (int t = 0; t < T; ++t) acc = fmaf(a[b * T + t], enc_out[(size_t)t * HD + hh], acc);
    ctx[i] = acc;
}

// cc[b][j] = tanh( concat(rnn[b],ctx[b]) . Wc[j] + bc[j] ), wave per (b,j)
__global__ void k_cc(const float* __restrict__ rnn, const float* __restrict__ ctx,
                     const float* __restrict__ Wc, const float* __restrict__ bc,
                     float* __restrict__ cc) {
    int wid = (blockIdx.x * blockDim.x + threadIdx.x) >> 5;
    int lane = threadIdx.x & 31;
    if (wid >= BW * HD) return;
    int b = wid >> 9, j = wid & (HD - 1);
    const float* w = Wc + (size_t)j * (2 * HD);
    float acc = 0.f;
    for (int k = lane; k < HD; k += 32) acc = fmaf(rnn[(size_t)b * HD + k], w[k], acc);
    for (int k = lane; k < HD; k += 32) acc = fmaf(ctx[(size_t)b * HD + k], w[HD + k], acc);
    acc = wave_sum(acc);
    if (lane == 0) cc[(size_t)b * HD + j] = tanhf(acc + bc[j]);
}

// Output projection via f32 WMMA: logits[b][v] = cc[b] . Wout[v] + bout[v].
// One wave per 16-vocab tile; M padded 5->16 (zero rows in LDS), K=512 in
// 128 chained V_WMMA_F32_16X16X4_F32 ops. EXEC is all-ones around every WMMA.
__global__ void __launch_bounds__(256) k_logits(const float* __restrict__ cc,
                                                const float* __restrict__ Wout,
                                                const float* __restrict__ bout,
                                                float* __restrict__ logits) {
    __shared__ float sc[16 * (HD + 1)];   // stride 513 -> conflict-free A reads
    int tid = threadIdx.x;
    for (int i = tid; i < 16 * HD; i += 256) {
        int r = i / HD, c = i - r * HD;
        sc[r * (HD + 1) + c] = (r < BW) ? cc[r * HD + c] : 0.0f;
    }
    __syncthreads();
    int wave = tid >> 5, lane = tid & 31;
    int tile = blockIdx.x * 8 + wave;     // 2000 tiles total
    int n0 = tile * 16;
    int m = lane & 15;
    int ksel = (lane < 16) ? 0 : 2;       // 32-bit A/B frag layout: halves hold K+0/1 vs K+2/3
    const float* arow = sc + m * (HD + 1) + ksel;
    const float* wrow = Wout + (size_t)(n0 + m) * HD + ksel;
    v8f acc = {};
#pragma unroll 4
    for (int kb = 0; kb < HD / 4; ++kb) {
        int k = 4 * kb;
        v2f a, bf;
        a.x = arow[k];  a.y = arow[k + 1];
        bf.x = wrow[k]; bf.y = wrow[k + 1];
        acc = __builtin_amdgcn_wmma_f32_16x16x4_f32(false, a, false, bf,
                                                    (short)0, acc, false, false);
    }
    if (lane < 16) {
        float bb = bout[n0 + lane];
#pragma unroll
        for (int r = 0; r < BW; ++r)       // D: VGPR r / lanes 0-15 == row M=r
            logits[(size_t)r * VSZ + n0 + lane] = acc[r] + bb;
    }
}

// Per-beam max & log-sum-exp over 32000 (for log_softmax values of top-k).
__global__ void k_lse(const float* __restrict__ logits, float* __restrict__ red) {
    int b = blockIdx.x, tid = threadIdx.x;
    __shared__ float sm[256];
    const float* row = logits + (size_t)b * VSZ;
    float mx = -INFINITY;
    for (int i = tid; i < VSZ; i += 256) mx = fmaxf(mx, row[i]);
    sm[tid] = mx; __syncthreads();
    for (int s = 128; s > 0; s >>= 1) { if (tid < s) sm[tid] = fmaxf(sm[tid], sm[tid + s]); __syncthreads(); }
    float gmx = sm[0]; __syncthreads();
    float sum = 0.f;
    for (int i = tid; i < VSZ; i += 256) sum += expf(row[i] - gmx);
    sm[tid] = sum; __syncthreads();
    for (int s = 128; s > 0; s >>= 1) { if (tid < s) sm[tid] += sm[tid + s]; __syncthreads(); }
    if (tid == 0) { red[2 * b] = gmx; red[2 * b + 1] = logf(sm[0]); }
}

__device__ __forceinline__ bool tk_better(float v1, int i1, float v2, int i2) {
    return (v1 > v2) || ((v1 == v2) && ((unsigned)i1 < (unsigned)i2)); // lax.top_k tie rule
}

// Per-beam top-5 (monotonic under log_softmax, so done on raw logits).
__global__ void k_top5(const float* __restrict__ logits, float* __restrict__ tv,
                       int* __restrict__ ti) {
    int b = blockIdx.x, tid = threadIdx.x;
    __shared__ float sv[256 * 5];
    __shared__ int   si[256 * 5];
    const float* row = logits + (size_t)b * VSZ;
    float v[5]; int id[5];
#pragma unroll
    for (int r = 0; r < 5; ++r) { v[r] = -INFINITY; id[r] = 0x7fffffff; }
    for (int i = tid; i < VSZ; i += 256) {
        float x = row[i];
        if (tk_better(x, i, v[4], id[4])) {
            v[4] = x; id[4] = i;
            for (int r = 4; r > 0 && tk_better(v[r], id[r], v[r - 1], id[r - 1]); --r) {
                float tvv = v[r]; v[r] = v[r - 1]; v[r - 1] = tvv;
                int tii = id[r]; id[r] = id[r - 1]; id[r - 1] = tii;
            }
        }
    }
#pragma unroll
    for (int r = 0; r < 5; ++r) { sv[tid * 5 + r] = v[r]; si[tid * 5 + r] = id[r]; }
    __syncthreads();
    for (int s = 128; s > 0; s >>= 1) {
        if (tid < s) {
            for (int r = 0; r < 5; ++r) {
                float ov = sv[(tid + s) * 5 + r]; int oi = si[(tid + s) * 5 + r];
                if (tk_better(ov, oi, v[4], id[4])) {
                    v[4] = ov; id[4] = oi;
                    for (int q = 4; q > 0 && tk_better(v[q], id[q], v[q - 1], id[q - 1]); --q) {
                        float tvv = v[q]; v[q] = v[q - 1]; v[q - 1] = tvv;
                        int tii = id[q]; id[q] = id[q - 1]; id[q - 1] = tii;
                    }
                }
            }
            for (int r = 0; r < 5; ++r) { sv[tid * 5 + r] = v[r]; si[tid * 5 + r] = id[r]; }
        }
        __syncthreads();
    }
    if (tid == 0)
        for (int r = 0; r < 5; ++r) { tv[b * 5 + r] = v[r]; ti[b * 5 + r] = id[r]; }
}

// Beam combine: 25 candidates -> top-5 (tie: lowest flat index == b-major).
__global__ void k_combine(const float* __restrict__ tv, const int* __restrict__ ti,
                          const float* __restrict__ red, float* __restrict__ scores,
                          int* __restrict__ tokens, int* __restrict__ parent,
                          int* __restrict__ toks_h, int* __restrict__ pars_h, int s) {
    if (threadIdx.x != 0 || blockIdx.x != 0) return;
    float sc0[5];
    for (int b = 0; b < 5; ++b) sc0[b] = scores[b];
    float cand[25];
    for (int b = 0; b < 5; ++b)
        for (int i = 0; i < 5; ++i)
            cand[b * 5 + i] = sc0[b] + (tv[b * 5 + i] - red[2 * b] - red[2 * b + 1]);
    bool used[25] = {};
    float ns[5]; int nt[5], np[5];
    for (int r = 0; r < 5; ++r) {
        int best = 0; float bv = -INFINITY;
        for (int c = 0; c < 25; ++c)
            if (!used[c] && cand[c] > bv) { bv = cand[c]; best = c; }
        used[best] = true;
        np[r] = best / 5; nt[r] = ti[best]; ns[r] = bv;
    }
    for (int r = 0; r < 5; ++r) {
        scores[r] = ns[r]; tokens[r] = nt[r]; parent[r] = np[r];
        toks_h[s * 5 + r] = nt[r]; pars_h[s * 5 + r] = np[r];
    }
}

__global__ void k_initdec(const float* __restrict__ hfin, const int* __restrict__ sos,
                          float* __restrict__ hd, float* __restrict__ scores,
                          int* __restrict__ tokens, int* __restrict__ parent) {
    int i = blockIdx.x * blockDim.x + threadIdx.x;
    if (i < LD * BW * HD) {
        int l = i / (BW * HD);
        int h = i % HD;
        hd[i] = hfin[l * HD + h];
    }
    if (i < BW) { scores[i] = (i == 0) ? 0.0f : NEGF; tokens[i] = *sos; parent[i] = i; }
}

__global__ void k_back(const float* __restrict__ scores, const int* __restrict__ toks_h,
                       const int* __restrict__ pars_h, float* __restrict__ out) {
    if (threadIdx.x != 0 || blockIdx.x != 0) return;
    int best = 0; float bv = scores[0];
    for (int b = 1; b < 5; ++b) if (scores[b] > bv) { bv = scores[b]; best = b; }
    int beam = best;
    for (int s = TDEC - 1; s >= 0; --s) {
        out[s] = (float)toks_h[s * 5 + beam];
        beam = pars_h[s * 5 + beam];
    }
    out[TDEC] = bv;
}

extern "C" void kernel_launch(void* const* d_in, const int* in_sizes, int n_in,
                              void* d_out, int out_size, void* d_ws, size_t ws_size,
                              hipStream_t stream) {
    const int*   seq  = (const int*)d_in[0];
    const int*   sos  = (const int*)d_in[3];
    const float* emb  = (const float*)d_in[4];
    const float* eWi  = (const float*)d_in[5];
    const float* eWh  = (const float*)d_in[6];
    const float* ebi  = (const float*)d_in[7];
    const float* ebh  = (const float*)d_in[8];
    const float* dWi  = (const float*)d_in[9];
    const float* dWh  = (const float*)d_in[10];
    const float* dbi  = (const float*)d_in[11];
    const float* dbh  = (const float*)d_in[12];
    const float* Wc   = (const float*)d_in[13];
    const float* bc   = (const float*)d_in[14];
    const float* Wout = (const float*)d_in[15];
    const float* boutp= (const float*)d_in[16];
    const int T = in_sizes[0];            // encoder length (128)

    // workspace layout (all 4-byte typed)
    float* f       = (float*)d_ws;
    float* hA      = f;                               // LD*HD
    float* hB      = hA + LD * HD;                    // LD*HD
    float* enc_out = hB + LD * HD;                    // T*HD
    float* hd0     = enc_out + (size_t)T * HD;        // LD*BW*HD
    float* hd1     = hd0 + LD * BW * HD;              // LD*BW*HD
    float* e       = hd1 + LD * BW * HD;              // BW*T
    float* ctx     = e + BW * T;                      // BW*HD
    float* ccb     = ctx + BW * HD;                   // BW*HD
    float* logits  = ccb + BW * HD;                   // BW*VSZ
    float* red     = logits + (size_t)BW * VSZ;       // 2*BW
    float* tv      = red + 2 * BW;                    // 25
    float* scores  = tv + BW * BW;                    // 5
    int*   ti      = (int*)(scores + BW);             // 25
    int*   parent  = ti + BW * BW;                    // 5
    int*   tokens  = parent + BW;                     // 5
    int*   toks_h  = tokens + BW;                     // TDEC*BW
    int*   pars_h  = toks_h + TDEC * BW;              // TDEC*BW
    float* hd[2] = { hd0, hd1 };

    // h0 = zeros
    k_zero<<<4, 256, 0, stream>>>(hA, LD * HD);

    // ---- encoder: 2T dependent fused-GRU launches, ping-pong hA/hB ----
    for (int t = 0; t < T; ++t) {
        float* hr = (t & 1) ? hB : hA;
        float* hw = (t & 1) ? hA : hB;
        // layer 0: x = emb[seq[t]]
        k_gru<<<64, 256, 0, stream>>>(emb, seq + t, hr, nullptr,
                                      eWi, eWh, ebi, ebh, hw, nullptr, 1);
        // layer 1: x = layer-0 new hidden; also copy to enc_out[t]
        k_gru<<<64, 256, 0, stream>>>(hw, nullptr, hr + HD, nullptr,
                                      eWi + 3 * HD * HD, eWh + 3 * HD * HD,
                                      ebi + 3 * HD, ebh + 3 * HD,
                                      hw + HD, enc_out + (size_t)t * HD, 1);
    }
    float* hfin = (T & 1) ? hB : hA;

    // ---- decoder init: broadcast final hidden, scores=[0,NEG..], tokens=SOS ----
    k_initdec<<<20, 256, 0, stream>>>(hfin, sos, hd[0], scores, tokens, parent);

    // ---- beam-search decode ----
    for (int s = 0; s < TDEC; ++s) {
        float* Hr = hd[s & 1];
        float* Hw = hd[(s + 1) & 1];
        // layer 0: x = emb[tokens[b]], h gathered by parent
        k_gru<<<320, 256, 0, stream>>>(emb, tokens, Hr, parent,
                                       dWi, dWh, dbi, dbh, Hw, nullptr, BW);
        // layer 1: x = layer-0 output (per beam), h gathered by parent
        k_gru<<<320, 256, 0, stream>>>(Hw, nullptr, Hr + BW * HD, parent,
                                       dWi + 3 * HD * HD, dWh + 3 * HD * HD,
                                       dbi + 3 * HD, dbh + 3 * HD,
                                       Hw + BW * HD, nullptr, BW);
        float* rnn = Hw + BW * HD;
        k_attn<<<(BW * T * 32 + 255) / 256, 256, 0, stream>>>(rnn, enc_out, e, T);
        k_softmax<<<BW, T, 0, stream>>>(e, T);
        k_ctx<<<(BW * HD + 255) / 256, 256, 0, stream>>>(e, enc_out, ctx, T);
        k_cc<<<320, 256, 0, stream>>>(rnn, ctx, Wc, bc, ccb);
        k_logits<<<VSZ / 16 / 8, 256, 0, stream>>>(ccb, Wout, boutp, logits);
        k_lse<<<BW, 256, 0, stream>>>(logits, red);
        k_top5<<<BW, 256, 0, stream>>>(logits, tv, ti);
        k_combine<<<1, 32, 0, stream>>>(tv, ti, red, scores, tokens, parent,
                                        toks_h, pars_h, s);
    }

    // ---- backtrack best beam -> d_out[0..63] tokens, d_out[64] score ----
    k_back<<<1, 1, 0, stream>>>(scores, toks_h, pars_h, (float*)d_out);
}